// GAT_20779051778181
// MI455X (gfx1250) — compile-verified
//
#include <hip/hip_runtime.h>
#include <hip/hip_bf16.h>
#include <math.h>

// ---------------------------------------------------------------------------
// GAT (2x GATConv + 2 MLP heads) for MI455X / gfx1250, wave32.
//
// Perf model (MI455X): whole net is ~100 MB unique data -> HBM roofline ~5us.
// feat [N,128] f32 = 25.6 MB fits in the 192 MB L2, so the 1 GB/layer of
// random edge gathers and the 128M f32 atomic adds/layer are L2-resident.
// GEMMs (~3.7 GFLOP total) are <1% of runtime -> exact fp32 via
// V_WMMA_F32_16X16X4_F32. Each wave computes a 16x64 strip so one A fragment
// feeds 4 WMMAs; main K-loop is branch-free (tail handled once, outside).
// ---------------------------------------------------------------------------

#define NN     50000
#define EE     1000000
#define F_IN   165
#define HEADS  4
#define F_OUT  32
#define HID    128
#define NEG_SLOPE 0.2f
#define NEG_INF_INIT (-1.0e30f)

typedef __attribute__((ext_vector_type(2))) float v2f;
typedef __attribute__((ext_vector_type(8))) float v8f;

// ---------------------------------------------------------------------------
// fp32 WMMA GEMM: C[M x 128] = A[M x K] * B[K x 128], row-major everything.
// One wave computes a 16x64 strip of C (4 16x16 WMMA tiles along N).
// A 16x4 f32 fragment layout (ISA 7.12.2): lanes 0-15 -> M=lane, K={k0,k0+1};
// lanes 16-31 -> M=lane-16, K={k0+2,k0+3}. B 4x16 symmetric.
// C/D: VGPR v, lanes 0-15 -> M=v, N=lane; lanes 16-31 -> M=v+8, N=lane-16.
// ---------------------------------------------------------------------------
__global__ void wmma_gemm_f32(const float* __restrict__ A,
                              const float* __restrict__ B,
                              float* __restrict__ C,
                              int M, int K) {
  const int stripsN = HID / 64;                      // 2 strips along N=128
  int wave = blockIdx.x * (blockDim.x >> 5) + (threadIdx.x >> 5);
  int tm = wave / stripsN;
  int tn = (wave % stripsN) * 64;
  if (tm * 16 >= M) return;

  int lane  = threadIdx.x & 31;
  int r     = lane & 15;                             // row (A) / col (B,C)
  int khalf = (lane >> 4) * 2;                       // 0 or 2

  const float* Ap = A + (size_t)(tm * 16 + r) * K + khalf;   // A[row, khalf]
  const float* Bp = B + (size_t)khalf * HID + tn + r;        // B[khalf, col]

  v8f acc0 = {}, acc1 = {}, acc2 = {}, acc3 = {};

  int Kmain = K & ~3;
  int k0 = 0;
  // branch-free main loop: 1 A frag -> 4 WMMAs
  for (; k0 < Kmain; k0 += 4) {
    v2f a;
    a.x = Ap[0];
    a.y = Ap[1];
    v2f b0, b1, b2, b3;
    b0.x = Bp[0];   b0.y = Bp[HID];
    b1.x = Bp[16];  b1.y = Bp[16 + HID];
    b2.x = Bp[32];  b2.y = Bp[32 + HID];
    b3.x = Bp[48];  b3.y = Bp[48 + HID];
    acc0 = __builtin_amdgcn_wmma_f32_16x16x4_f32(false, a, false, b0,
                                                 (short)0, acc0, false, false);
    acc1 = __builtin_amdgcn_wmma_f32_16x16x4_f32(false, a, false, b1,
                                                 (short)0, acc1, false, false);
    acc2 = __builtin_amdgcn_wmma_f32_16x16x4_f32(false, a, false, b2,
                                                 (short)0, acc2, false, false);
    acc3 = __builtin_amdgcn_wmma_f32_16x16x4_f32(false, a, false, b3,
                                                 (short)0, acc3, false, false);
    Ap += 4;
    Bp += 4 * HID;
  }
  // K tail (K % 4 != 0): runs at most once, guards allowed here
  if (k0 < K) {
    int ka = k0 + khalf;
    v2f a;
    a.x = (ka     < K) ? Ap[0] : 0.0f;
    a.y = (ka + 1 < K) ? Ap[1] : 0.0f;
    v2f b0 = {}, b1 = {}, b2 = {}, b3 = {};
    if (ka < K) {
      b0.x = Bp[0];  b1.x = Bp[16];  b2.x = Bp[32];  b3.x = Bp[48];
    }
    if (ka + 1 < K) {
      b0.y = Bp[HID];       b1.y = Bp[16 + HID];
      b2.y = Bp[32 + HID];  b3.y = Bp[48 + HID];
    }
    acc0 = __builtin_amdgcn_wmma_f32_16x16x4_f32(false, a, false, b0,
                                                 (short)0, acc0, false, false);
    acc1 = __builtin_amdgcn_wmma_f32_16x16x4_f32(false, a, false, b1,
                                                 (short)0, acc1, false, false);
    acc2 = __builtin_amdgcn_wmma_f32_16x16x4_f32(false, a, false, b2,
                                                 (short)0, acc2, false, false);
    acc3 = __builtin_amdgcn_wmma_f32_16x16x4_f32(false, a, false, b3,
                                                 (short)0, acc3, false, false);
  }

  // store: VGPR v -> row (v + 8*(lane>=16)), col r within each 16-wide tile
  int rbase = (lane >> 4) * 8;
  float* Crow = C + (size_t)(tm * 16) * HID + tn + r;
#pragma unroll
  for (int v = 0; v < 8; ++v) {
    size_t ro = (size_t)(rbase + v) * HID;
    Crow[ro]      = acc0[v];
    Crow[ro + 16] = acc1[v];
    Crow[ro + 32] = acc2[v];
    Crow[ro + 48] = acc3[v];
  }
}

// ---------------------------------------------------------------------------
// Per-node attention logits: el[n,h] = sum_f feat[n,h*32+f]*al[h,f]; er same.
// Block = 128 threads = 4 wave32s = 4 heads; shuffle-reduce within each wave.
// ---------------------------------------------------------------------------
__global__ void node_logits(const float* __restrict__ feat,
                            const float* __restrict__ al,
                            const float* __restrict__ ar,
                            float* __restrict__ el,
                            float* __restrict__ er) {
  int n = blockIdx.x;
  int t = threadIdx.x;          // 0..127
  int h = t >> 5;
  int f = t & 31;
  float v  = feat[(size_t)n * HID + t];
  float pl = v * al[h * F_OUT + f];
  float pr = v * ar[h * F_OUT + f];
#pragma unroll
  for (int off = 16; off > 0; off >>= 1) {
    pl += __shfl_down(pl, off, 32);
    pr += __shfl_down(pr, off, 32);
  }
  if (f == 0) {
    el[n * HEADS + h] = pl;
    er[n * HEADS + h] = pr;
  }
}

__global__ void fill_f32(float* __restrict__ p, float val, int n) {
  for (int i = blockIdx.x * blockDim.x + threadIdx.x; i < n;
       i += gridDim.x * blockDim.x)
    p[i] = val;
}

__device__ __forceinline__ float leaky(float x) {
  return x > 0.0f ? x : NEG_SLOPE * x;
}

// float atomic max via signed/unsigned integer ordering trick
__device__ __forceinline__ void atomicMaxF(float* addr, float val) {
  if (!(val < 0.0f))
    atomicMax((int*)addr, __float_as_int(val));
  else
    atomicMin((unsigned int*)addr, (unsigned int)__float_as_int(val));
}

// ---------------------------------------------------------------------------
// Edge pass 1: segment max of leaky_relu(el[src]+er[dst]) into emax[dst].
// ---------------------------------------------------------------------------
__global__ void edge_max(const int* __restrict__ src,
                         const int* __restrict__ dst,
                         const float* __restrict__ el,
                         const float* __restrict__ er,
                         float* __restrict__ emax) {
  int e = blockIdx.x * blockDim.x + threadIdx.x;
  if (e >= EE) return;
  int s = src[e], d = dst[e];
  const float4 l = *(const float4*)(el + (size_t)s * HEADS);
  const float4 r = *(const float4*)(er + (size_t)d * HEADS);
  float z0 = leaky(l.x + r.x), z1 = leaky(l.y + r.y);
  float z2 = leaky(l.z + r.z), z3 = leaky(l.w + r.w);
  float* em = emax + (size_t)d * HEADS;
  atomicMaxF(em + 0, z0);
  atomicMaxF(em + 1, z1);
  atomicMaxF(em + 2, z2);
  atomicMaxF(em + 3, z3);
}

// isolated-node guard: reference maps non-finite segment max to 0
__global__ void emax_fix(float* __restrict__ emax, int n) {
  int i = blockIdx.x * blockDim.x + threadIdx.x;
  if (i < n && emax[i] < -1.0e29f) emax[i] = 0.0f;
}

// ---------------------------------------------------------------------------
// Edge pass 2: ex = exp(leaky(el[src]+er[dst]) - emax[dst]); denom[dst] += ex.
// ---------------------------------------------------------------------------
__global__ void edge_expsum(const int* __restrict__ src,
                            const int* __restrict__ dst,
                            const float* __restrict__ el,
                            const float* __restrict__ er,
                            const float* __restrict__ emax,
                            float* __restrict__ ex,
                            float* __restrict__ denom) {
  int e = blockIdx.x * blockDim.x + threadIdx.x;
  if (e >= EE) return;
  int s = src[e], d = dst[e];
  const float4 l = *(const float4*)(el + (size_t)s * HEADS);
  const float4 r = *(const float4*)(er + (size_t)d * HEADS);
  const float4 m = *(const float4*)(emax + (size_t)d * HEADS);
  float x0 = expf(leaky(l.x + r.x) - m.x);
  float x1 = expf(leaky(l.y + r.y) - m.y);
  float x2 = expf(leaky(l.z + r.z) - m.z);
  float x3 = expf(leaky(l.w + r.w) - m.w);
  float4 xv = {x0, x1, x2, x3};
  *(float4*)(ex + (size_t)e * HEADS) = xv;
  float* dn = denom + (size_t)d * HEADS;
  atomicAdd(dn + 0, x0);
  atomicAdd(dn + 1, x1);
  atomicAdd(dn + 2, x2);
  atomicAdd(dn + 3, x3);
}

// ---------------------------------------------------------------------------
// Edge pass 3: out[dst, :] += feat[src, :] * alpha[e, head].
// 128 threads per edge (coalesced 512 B gather per edge); L2-resident atomics.
// NOTE: float atomic add order is unordered -> last-ulp nondeterminism across
// replays; acceptable (no deterministic alternative without an edge sort).
// ---------------------------------------------------------------------------
__global__ void edge_aggregate(const int* __restrict__ src,
                               const int* __restrict__ dst,
                               const float* __restrict__ feat,
                               const float* __restrict__ ex,
                               const float* __restrict__ denom,
                               float* __restrict__ out) {
  long long gid = (long long)blockIdx.x * blockDim.x + threadIdx.x;
  int e = (int)(gid >> 7);
  if (e >= EE) return;
  int t = (int)(gid & 127);
  int h = t >> 5;
  int s = src[e], d = dst[e];
  float alpha = ex[(size_t)e * HEADS + h] /
                fmaxf(denom[(size_t)d * HEADS + h], 1e-9f);
  atomicAdd(&out[(size_t)d * HID + t], feat[(size_t)s * HID + t] * alpha);
}

__global__ void relu_inplace(float* __restrict__ p, int n) {
  int i = blockIdx.x * blockDim.x + threadIdx.x;
  if (i < n) p[i] = fmaxf(p[i], 0.0f);
}

// ---------------------------------------------------------------------------
// Output heads: predict = relu(h@Wc1+bc1)@Wc2+bc2 ;
//               confidence = sigmoid(relu(h@Wf1+bf1)@Wf2+bf2).
// One wave32 per node; each lane owns 4 of the 128 hidden values.
// ---------------------------------------------------------------------------
__global__ void heads_kernel(const float* __restrict__ h,
                             const float* __restrict__ Wc1,
                             const float* __restrict__ bc1,
                             const float* __restrict__ Wc2,
                             const float* __restrict__ bc2,
                             const float* __restrict__ Wf1,
                             const float* __restrict__ bf1,
                             const float* __restrict__ Wf2,
                             const float* __restrict__ bf2,
                             float* __restrict__ out) {
  int node = blockIdx.x * (blockDim.x >> 5) + (threadIdx.x >> 5);
  if (node >= NN) return;
  int lane = threadIdx.x & 31;

  float hv[4];
#pragma unroll
  for (int i = 0; i < 4; ++i) hv[i] = h[(size_t)node * HID + lane + 32 * i];

  float zc[10], zf[10];
#pragma unroll
  for (int j = 0; j < 10; ++j) {
    float pc = 0.0f, pf = 0.0f;
#pragma unroll
    for (int i = 0; i < 4; ++i) {
      int t = lane + 32 * i;
      pc += hv[i] * Wc1[t * 10 + j];
      pf += hv[i] * Wf1[t * 10 + j];
    }
#pragma unroll
    for (int off = 16; off > 0; off >>= 1) {
      pc += __shfl_down(pc, off, 32);
      pf += __shfl_down(pf, off, 32);
    }
    zc[j] = pc;
    zf[j] = pf;
  }
  if (lane == 0) {
    float pc = bc2[0], pf = bf2[0];
#pragma unroll
    for (int j = 0; j < 10; ++j) {
      pc += fmaxf(zc[j] + bc1[j], 0.0f) * Wc2[j];
      pf += fmaxf(zf[j] + bf1[j], 0.0f) * Wf2[j];
    }
    out[node]      = pc;                               // predict
    out[NN + node] = 1.0f / (1.0f + expf(-pf));        // confidence
  }
}

// ---------------------------------------------------------------------------
// Workspace layout (floats). feat buffer reused across layers; h buffer is
// h1, then (after GEMM2 consumes it) the layer-2 scatter target = h2.
// Total: 17.6M floats = 70.4 MB.
// ---------------------------------------------------------------------------
#define OF_FEAT  0                                  // N*HID
#define OF_H     (OF_FEAT  + NN * HID)              // N*HID
#define OF_EX    (OF_H     + NN * HID)              // E*HEADS
#define OF_EL    (OF_EX    + EE * HEADS)            // N*HEADS
#define OF_ER    (OF_EL    + NN * HEADS)
#define OF_EMAX  (OF_ER    + NN * HEADS)
#define OF_DENOM (OF_EMAX  + NN * HEADS)

static inline void run_gat_layer(const float* x, int K,
                                 const float* W, const float* al,
                                 const float* ar,
                                 const int* src, const int* dst,
                                 float* feat, float* hout,
                                 float* el, float* er,
                                 float* emax, float* denom, float* ex,
                                 hipStream_t stream) {
  // GEMM: feat = x @ W   (M = NN rows, exact multiple of 16)
  int waves  = (NN / 16) * (HID / 64);               // 6250 strip-waves
  int blocks = (waves + 7) / 8;                      // 8 waves / 256-thr block
  wmma_gemm_f32<<<blocks, 256, 0, stream>>>(x, W, feat, NN, K);

  node_logits<<<NN, HID, 0, stream>>>(feat, al, ar, el, er);

  fill_f32<<<512, 256, 0, stream>>>(emax, NEG_INF_INIT, NN * HEADS);
  fill_f32<<<512, 256, 0, stream>>>(denom, 0.0f, NN * HEADS);
  fill_f32<<<2048, 256, 0, stream>>>(hout, 0.0f, NN * HID);

  int eb = (EE + 255) / 256;
  edge_max<<<eb, 256, 0, stream>>>(src, dst, el, er, emax);
  emax_fix<<<(NN * HEADS + 255) / 256, 256, 0, stream>>>(emax, NN * HEADS);
  edge_expsum<<<eb, 256, 0, stream>>>(src, dst, el, er, emax, ex, denom);

  long long tot = (long long)EE * HID;
  int ab = (int)((tot + 255) / 256);                 // 500000 blocks
  edge_aggregate<<<ab, 256, 0, stream>>>(src, dst, feat, ex, denom, hout);

  relu_inplace<<<(NN * HID + 255) / 256, 256, 0, stream>>>(hout, NN * HID);
}

extern "C" void kernel_launch(void* const* d_in, const int* in_sizes, int n_in,
                              void* d_out, int out_size, void* d_ws,
                              size_t ws_size, hipStream_t stream) {
  const float* features = (const float*)d_in[0];
  const int*   src      = (const int*)d_in[1];
  const int*   dst      = (const int*)d_in[2];
  const float* W1  = (const float*)d_in[3];
  const float* al1 = (const float*)d_in[4];
  const float* ar1 = (const float*)d_in[5];
  const float* W2  = (const float*)d_in[6];
  const float* al2 = (const float*)d_in[7];
  const float* ar2 = (const float*)d_in[8];
  const float* Wc1 = (const float*)d_in[9];
  const float* bc1 = (const float*)d_in[10];
  const float* Wc2 = (const float*)d_in[11];
  const float* bc2 = (const float*)d_in[12];
  const float* Wf1 = (const float*)d_in[13];
  const float* bf1 = (const float*)d_in[14];
  const float* Wf2 = (const float*)d_in[15];
  const float* bf2 = (const float*)d_in[16];

  float* ws    = (float*)d_ws;
  float* feat  = ws + OF_FEAT;
  float* hbuf  = ws + OF_H;
  float* ex    = ws + OF_EX;
  float* el    = ws + OF_EL;
  float* er    = ws + OF_ER;
  float* emax  = ws + OF_EMAX;
  float* denom = ws + OF_DENOM;
  float* out   = (float*)d_out;

  // Layer 1: features[N,165] -> hbuf[N,128]
  run_gat_layer(features, F_IN, W1, al1, ar1, src, dst,
                feat, hbuf, el, er, emax, denom, ex, stream);

  // Layer 2: hbuf[N,128] -> hbuf[N,128] (GEMM2 consumes hbuf into feat first,
  // then hbuf is zeroed and becomes the layer-2 scatter target)
  run_gat_layer(hbuf, HID, W2, al2, ar2, src, dst,
                feat, hbuf, el, er, emax, denom, ex, stream);

  // Heads -> d_out = [predict(N) | confidence(N)]
  heads_kernel<<<(NN + 7) / 8, 256, 0, stream>>>(
      hbuf, Wc1, bc1, Wc2, bc2, Wf1, bf1, Wf2, bf2, out);
}